// _SocialPooling_32469952758508
// MI455X (gfx1250) — compile-verified
//
#include <hip/hip_runtime.h>
#include <hip/hip_bf16.h>

// ---------------- problem constants (match reference) ----------------
#define HD   1024      // hidden dim H
#define GSZ  8         // grid size
#define NBR  4.0f      // neighborhood radius (L-inf)

typedef float v2f __attribute__((ext_vector_type(2)));
typedef float v8f __attribute__((ext_vector_type(8)));

// cell id from relative position (matches jnp.clip + floor semantics; val >= 0
// so int truncation == floor)
__device__ __forceinline__ int sp_cell(float dx, float dy) {
    float vx = fminf(fmaxf((dx / NBR + 1.0f) * 0.5f * (float)GSZ, 0.0f), (float)(GSZ - 1));
    float vy = fminf(fmaxf((dy / NBR + 1.0f) * 0.5f * (float)GSZ, 0.0f), (float)(GSZ - 1));
    return (int)vy * GSZ + (int)vx;
}

// ---------------- kernel 1: zero workspace accumulators ----------------
__global__ void sp_init(int* __restrict__ counts, float* __restrict__ pooled) {
    int t = blockIdx.x * blockDim.x + threadIdx.x;
    if (t < GSZ * GSZ) counts[t] = 0;
    if (t < HD)        pooled[t] = 0.0f;
}

// ---------------- kernel 2: per-cell neighbor counts ----------------
__global__ void __launch_bounds__(256)
sp_count(const float* __restrict__ pos, const int* __restrict__ idxp,
         int* __restrict__ counts, int n) {
    int a = blockIdx.x * blockDim.x + threadIdx.x;
    if (a >= n) return;
    int idx = idxp[0];
    float px = pos[2 * idx], py = pos[2 * idx + 1];
    float dx = pos[2 * a] - px, dy = pos[2 * a + 1] - py;
    if (a != idx && fmaxf(fabsf(dx), fabsf(dy)) <= NBR) {
        atomicAdd(&counts[sp_cell(dx, dy)], 1);
    }
}

// ---------------- kernel 3: weighted row-sum (pooled vector) ----------------
// pooled = sum_i mask_i * h_i / count[cell_i]   (identical to the reference's
// segment_sum -> divide -> sum-over-cells, fused).
// Each block handles 256 agents. Weights are staged in LDS so the w!=0 test
// is block-uniform: masked-out rows (~84% of agents) generate ZERO h_states
// traffic — the kernel moves only ~66 MB instead of 410 MB.
__global__ void __launch_bounds__(256)
sp_pool(const float* __restrict__ h, const float* __restrict__ pos,
        const int* __restrict__ idxp, const int* __restrict__ counts,
        float* __restrict__ pooled, int n) {
    __shared__ float wsh[256];
    const int tid  = threadIdx.x;
    const int base = blockIdx.x * 256;
    const int a    = base + tid;

    const int idx = idxp[0];
    const float px = pos[2 * idx], py = pos[2 * idx + 1];

    float w = 0.0f;
    if (a < n && a != idx) {
        float dx = pos[2 * a] - px, dy = pos[2 * a + 1] - py;
        if (fmaxf(fabsf(dx), fabsf(dy)) <= NBR) {
            int c = counts[sp_cell(dx, dy)];
            if (c < 1) c = 1;
            w = 1.0f / (float)c;
        }
    }
    wsh[tid] = w;
    __syncthreads();

    float acc0 = 0.0f, acc1 = 0.0f, acc2 = 0.0f, acc3 = 0.0f;
    const int limit = min(256, n - base);
    for (int j = 0; j < limit; ++j) {
        const float wj = wsh[j];           // uniform across the block
        if (wj != 0.0f) {                  // whole wave skips dead rows
            const float* row = h + (size_t)(base + j) * HD;
            acc0 += row[tid        ] * wj; // coalesced 4 KB row read
            acc1 += row[tid +  256 ] * wj;
            acc2 += row[tid +  512 ] * wj;
            acc3 += row[tid +  768 ] * wj;
        }
    }
    atomicAdd(&pooled[tid       ], acc0);
    atomicAdd(&pooled[tid +  256], acc1);
    atomicAdd(&pooled[tid +  512], acc2);
    atomicAdd(&pooled[tid +  768], acc3);
}

// ---------------- kernel 4: out = W * pooled + b via V_WMMA_F32_16X16X4_F32 --
// D(16x16) = A(16x4) x B(4x16) + C, full f32 precision (matches reference).
//   A[m][k] = pooled[k0+k]      (broadcast over rows m)
//   B[k][n] = W[c0+n][k0+k]     (W row-major, out[j] = sum_k pooled[k]*W[j][k])
// => D[0][n] accumulates out[c0+n] across the K loop (C-chained).
// Per ISA layout: A lanes 0-15 carry K={0,1}, lanes 16-31 carry K={2,3};
// B VGPR0/1 = K rows {0,1} (lanes 0-15, N=lane) / {2,3} (lanes 16-31).
// D VGPR0, lanes 0-15 = row M=0, N=lane.
__global__ void __launch_bounds__(32)
sp_gemv_wmma(const float* __restrict__ W, const float* __restrict__ b,
             const float* __restrict__ pooled, float* __restrict__ out) {
    const int c0    = blockIdx.x * 16;      // 16 output elements per wave
    const int lane  = threadIdx.x;          // wave32
    const int l     = lane & 15;
    const int khalf = (lane >> 4) * 2;      // lanes 16-31 hold K+2,K+3

    const float* wrow = W + (size_t)(c0 + l) * HD + khalf;
    const float* prow = pooled + khalf;

    v8f acc = {};
#pragma unroll 4
    for (int k0 = 0; k0 < HD; k0 += 4) {
        v2f A = *(const v2f*)(prow + k0);   // pooled[k0+khalf .. +1]
        v2f B = *(const v2f*)(wrow + k0);   // W[c0+l][k0+khalf .. +1]
        acc = __builtin_amdgcn_wmma_f32_16x16x4_f32(
            /*neg_a=*/false, A, /*neg_b=*/false, B,
            /*c_mod=*/(short)0, acc, /*reuse_a=*/false, /*reuse_b=*/false);
    }
    if (lane < 16) out[c0 + l] = acc[0] + b[c0 + l];  // row 0 of D
}

// ---------------- launcher ----------------
extern "C" void kernel_launch(void* const* d_in, const int* in_sizes, int n_in,
                              void* d_out, int out_size, void* d_ws, size_t ws_size,
                              hipStream_t stream) {
    const float* h    = (const float*)d_in[0];   // (N, H)
    const float* pos  = (const float*)d_in[1];   // (N, 2)
    const int*   idxp = (const int*)  d_in[2];   // scalar idx
    const float* W    = (const float*)d_in[3];   // (H, H)
    const float* b    = (const float*)d_in[4];   // (H,)
    float*       out  = (float*)d_out;           // (H,) float32

    const int n = in_sizes[0] / HD;

    // workspace layout: [64 x int counts][1024 x float pooled]
    int*   counts = (int*)d_ws;
    float* pooled = (float*)((char*)d_ws + 64 * sizeof(int));

    sp_init<<<4, 256, 0, stream>>>(counts, pooled);

    const int nb = (n + 255) / 256;
    sp_count<<<nb, 256, 0, stream>>>(pos, idxp, counts, n);
    sp_pool <<<nb, 256, 0, stream>>>(h, pos, idxp, counts, pooled, n);

    sp_gemv_wmma<<<HD / 16, 32, 0, stream>>>(W, b, pooled, out);
}